// WeightModel_9337258902085
// MI455X (gfx1250) — compile-verified
//
#include <hip/hip_runtime.h>
#include <hip/hip_bf16.h>

typedef float v2f __attribute__((ext_vector_type(2)));
typedef float v8f __attribute__((ext_vector_type(8)));

// D = A(16x4 f32) * B(4x16 f32) + C(16x16 f32), wave32 WMMA.
// A layout: lane l (M = l&15) holds K = 2*(l>>4) + {0,1} in its 2 VGPRs.
// B layout (mirrored): lane l (N = l&15) holds K = 2*(l>>4) + {0,1}.
// C layout: lane l -> col (l&15); VGPR v -> row v + 8*(l>>4).
__device__ __forceinline__ v8f wmma4(v2f a, v2f b, v8f c) {
    return __builtin_amdgcn_wmma_f32_16x16x4_f32(
        /*neg_a=*/false, a, /*neg_b=*/false, b,
        /*c_mod=*/(short)0, c, /*reuse_a=*/false, /*reuse_b=*/false);
}

// Pack weight rows [0, 4*KS) of a row-major (rows x 64) matrix into LDS as
// k-pairs, laid out [ks][c][half] so the per-WMMA B fragment is ONE b64 DS
// read and the 32 lanes hit 32 consecutive 8B slots -> all 64 banks once.
__device__ __forceinline__ void pack_B(v2f* sW2, const float* __restrict__ W, int KS) {
    int total = KS * 128;                       // float2 elements
    for (int idx = threadIdx.x; idx < total; idx += blockDim.x) {
        int ks  = idx >> 7;
        int rem = idx & 127;
        int c   = rem >> 1;
        int h   = rem & 1;
        int k0  = ks * 4 + 2 * h;
        v2f p; p.x = W[k0 * 64 + c]; p.y = W[(k0 + 1) * 64 + c];
        sW2[idx] = p;
    }
}

// ---------------- degree / initial embedding ----------------
__global__ void k_zero_i(int* __restrict__ p, int n) {
    int i = blockIdx.x * blockDim.x + threadIdx.x;
    if (i < n) p[i] = 0;
}

__global__ void k_deg(const int* __restrict__ src, int* __restrict__ deg, int E) {
    int i = blockIdx.x * blockDim.x + threadIdx.x;
    if (i < E) atomicAdd(&deg[src[i]], 1);
}

__global__ void k_emb0(const int* __restrict__ deg, float* __restrict__ emb0,
                       int n, float inv) {
    int i = blockIdx.x * blockDim.x + threadIdx.x;
    if (i < n) emb0[i] = (float)deg[i] * inv;
}

// ---------------- round 1 edge matmul + segment max ----------------
// One wave per node. msg = relu([emb0[nbr], lf0, lf1, lf2] @ Wa0 + ba0); agg = rowmax.
// K = 4 exactly -> a single WMMA per 16-col tile.
__global__ void k_edge1(const float* __restrict__ emb0, const float* __restrict__ lf,
                        const int* __restrict__ nbr, const float* __restrict__ Wa0,
                        const float* __restrict__ ba0, float* __restrict__ agg,
                        int nnodes) {
    __shared__ v2f sW2[128];                   // 1 KB: Wa0 pair-packed (KS=1)
    pack_B(sW2, Wa0, 1);
    __syncthreads();
    int wave = threadIdx.x >> 5, lane = threadIdx.x & 31;
    int node = blockIdx.x * (blockDim.x >> 5) + wave;
    if (node >= nnodes) return;
    int r = lane & 15, half = lane >> 4;
    long e = (long)node * 16 + r;

    v2f a;
    if (half == 0) { a.x = emb0[nbr[e]];  a.y = lf[e * 3 + 0]; }
    else           { a.x = lf[e * 3 + 1]; a.y = lf[e * 3 + 2]; }

    v8f acc[4];
#pragma unroll
    for (int nt = 0; nt < 4; ++nt) {
        int c = nt * 16 + r;
        float bias = ba0[c];
#pragma unroll
        for (int v = 0; v < 8; ++v) acc[nt][v] = bias;
        acc[nt] = wmma4(a, sW2[c * 2 + half], acc[nt]);
    }

    float m[4];
#pragma unroll
    for (int nt = 0; nt < 4; ++nt) {
        float v = acc[nt][0];
#pragma unroll
        for (int k = 1; k < 8; ++k) v = fmaxf(v, acc[nt][k]);
        v = fmaxf(v, __shfl_xor(v, 16, 32));
        m[nt] = fmaxf(v, 0.0f);                // relu and max commute
    }
#pragma unroll
    for (int i = 0; i < 2; ++i) {              // full-EXEC stores, no divergence
        float val = half ? m[2 + i] : m[i];
        agg[(long)node * 64 + (2 * half + i) * 16 + r] = val;
    }
}

// ---------------- round 1 node update ----------------
// One wave per 16 nodes. emb1 = relu([agg | emb0] @ Wn0 + bn0), K = 65.
__global__ void k_node1(const float* __restrict__ agg, const float* __restrict__ emb0,
                        const float* __restrict__ Wn0, const float* __restrict__ bn0,
                        float* __restrict__ emb1, int nnodes) {
    __shared__ v2f sW2[16 * 128];              // 16 KB: Wn0 rows 0..63 pair-packed
    pack_B(sW2, Wn0, 16);
    __syncthreads();
    int wave = threadIdx.x >> 5, lane = threadIdx.x & 31;
    int tile = blockIdx.x * (blockDim.x >> 5) + wave;
    int nb = tile * 16;
    if (nb >= nnodes) return;
    int r = lane & 15, half = lane >> 4;
    const float* arow = agg + (long)(nb + r) * 64 + 2 * half;

    v8f acc[4];
#pragma unroll
    for (int nt = 0; nt < 4; ++nt) {
        int c = nt * 16 + r;
        float bias = bn0[c];
        float w64 = Wn0[64 * 64 + c];          // emb0 (K row 64) tail, L1-cached
#pragma unroll
        for (int v = 0; v < 8; ++v)
            acc[nt][v] = bias + emb0[nb + v + 8 * half] * w64;
    }
#pragma unroll
    for (int ks = 0; ks < 16; ++ks) {
        v2f a = *(const v2f*)(arow + ks * 4);
#pragma unroll
        for (int nt = 0; nt < 4; ++nt) {
            int c = nt * 16 + r;
            acc[nt] = wmma4(a, sW2[ks * 128 + c * 2 + half], acc[nt]);
        }
    }
#pragma unroll
    for (int nt = 0; nt < 4; ++nt) {
        int c = nt * 16 + r;
#pragma unroll
        for (int v = 0; v < 8; ++v)
            emb1[(long)(nb + v + 8 * half) * 64 + c] = fmaxf(acc[nt][v], 0.0f);
    }
}

// ---------------- round 2 edge matmul + segment max ----------------
// One wave per node. msg = relu([emb1[nbr] (64) | lf (3)] @ Wa1 + ba1); agg = rowmax.
// 16 WMMA K-steps over the gathered emb1 rows (L2-resident: 25.6MB << 192MB L2);
// lf/bias (K rows 64..66) folded into the C initializer with VALU.
__global__ void k_edge2(const float* __restrict__ emb1, const float* __restrict__ lf,
                        const int* __restrict__ nbr, const float* __restrict__ Wa1,
                        const float* __restrict__ ba1, float* __restrict__ agg,
                        int nnodes) {
    __shared__ v2f sW2[16 * 128];              // 16 KB: Wa1 rows 0..63 pair-packed
    __shared__ float sLF[8][16][3];            // per-wave link-feat tile
    pack_B(sW2, Wa1, 16);
    int wave = threadIdx.x >> 5, lane = threadIdx.x & 31;
    int node = blockIdx.x * (blockDim.x >> 5) + wave;
    if (node < nnodes && lane < 16) {
        long e = (long)node * 16 + lane;
        sLF[wave][lane][0] = lf[e * 3 + 0];
        sLF[wave][lane][1] = lf[e * 3 + 1];
        sLF[wave][lane][2] = lf[e * 3 + 2];
    }
    __syncthreads();
    if (node >= nnodes) return;

    int r = lane & 15, half = lane >> 4;
    long e0 = (long)node * 16;
    int src = nbr[e0 + r];
    const float* arow = emb1 + (long)src * 64 + 2 * half;

    v8f acc[4];
#pragma unroll
    for (int nt = 0; nt < 4; ++nt) {
        int c = nt * 16 + r;
        float bias = ba1[c];
        float w64 = Wa1[64 * 64 + c], w65 = Wa1[65 * 64 + c], w66 = Wa1[66 * 64 + c];
#pragma unroll
        for (int v = 0; v < 8; ++v) {
            int er = v + 8 * half;             // edge row of this C element
            acc[nt][v] = bias + sLF[wave][er][0] * w64
                              + sLF[wave][er][1] * w65
                              + sLF[wave][er][2] * w66;
        }
    }
#pragma unroll
    for (int ks = 0; ks < 16; ++ks) {          // full unroll: clause the 16 gathers
        v2f a = *(const v2f*)(arow + ks * 4);
#pragma unroll
        for (int nt = 0; nt < 4; ++nt) {
            int c = nt * 16 + r;
            acc[nt] = wmma4(a, sW2[ks * 128 + c * 2 + half], acc[nt]);
        }
    }
    float m[4];
#pragma unroll
    for (int nt = 0; nt < 4; ++nt) {
        float v = acc[nt][0];
#pragma unroll
        for (int k = 1; k < 8; ++k) v = fmaxf(v, acc[nt][k]);
        v = fmaxf(v, __shfl_xor(v, 16, 32));
        m[nt] = fmaxf(v, 0.0f);
    }
#pragma unroll
    for (int i = 0; i < 2; ++i) {
        float val = half ? m[2 + i] : m[i];
        agg[(long)node * 64 + (2 * half + i) * 16 + r] = val;
    }
}

// ---------------- round 2 node update ----------------
// One wave per 16 nodes. emb2 = relu([agg (64) | emb1 (64)] @ Wn1 + bn1), K = 128.
__global__ void k_node2(const float* __restrict__ agg, const float* __restrict__ emb1,
                        const float* __restrict__ Wn1, const float* __restrict__ bn1,
                        float* __restrict__ emb2, int nnodes) {
    __shared__ v2f sW2[32 * 128];              // 32 KB: Wn1 pair-packed
    pack_B(sW2, Wn1, 32);
    __syncthreads();
    int wave = threadIdx.x >> 5, lane = threadIdx.x & 31;
    int tile = blockIdx.x * (blockDim.x >> 5) + wave;
    int nb = tile * 16;
    if (nb >= nnodes) return;
    int r = lane & 15, half = lane >> 4;
    const float* a0 = agg  + (long)(nb + r) * 64 + 2 * half;
    const float* a1 = emb1 + (long)(nb + r) * 64 + 2 * half;

    v8f acc[4];
#pragma unroll
    for (int nt = 0; nt < 4; ++nt) {
        int c = nt * 16 + r;
        float bias = bn1[c];
#pragma unroll
        for (int v = 0; v < 8; ++v) acc[nt][v] = bias;
    }
#pragma unroll 8
    for (int ks = 0; ks < 32; ++ks) {
        const float* ap = (ks < 16) ? a0 : a1;
        v2f a = *(const v2f*)(ap + (ks & 15) * 4);
#pragma unroll
        for (int nt = 0; nt < 4; ++nt) {
            int c = nt * 16 + r;
            acc[nt] = wmma4(a, sW2[ks * 128 + c * 2 + half], acc[nt]);
        }
    }
#pragma unroll
    for (int nt = 0; nt < 4; ++nt) {
        int c = nt * 16 + r;
#pragma unroll
        for (int v = 0; v < 8; ++v)
            emb2[(long)(nb + v + 8 * half) * 64 + c] = fmaxf(acc[nt][v], 0.0f);
    }
}

// ---------------- output logits ----------------
// logits[t] = [emb2[entry_node[t]] | entry_feats[t]] @ Wo + bo.
// 16 lanes cooperate per entry so the 410MB of emb2 row traffic (L2-resident)
// is read as contiguous 64B segments; dot reduced with a 16-wide shfl tree.
__global__ void k_out(const float* __restrict__ emb2, const float* __restrict__ ef,
                      const int* __restrict__ enode, const float* __restrict__ Wo,
                      const float* __restrict__ bo, float* __restrict__ out, int T) {
    __shared__ float sWo[70];
    __shared__ float sb;
    if (threadIdx.x < 70) sWo[threadIdx.x] = Wo[threadIdx.x];
    if (threadIdx.x == 0) sb = bo[0];
    __syncthreads();
    int lane = threadIdx.x & 31;
    int j = lane & 15;                          // lane within entry group
    int g = lane >> 4;                          // entry group within wave
    long wv = (long)blockIdx.x * (blockDim.x >> 5) + (threadIdx.x >> 5);
    long t = wv * 2 + g;
    bool ok = t < T;
    long tt = ok ? t : 0;
    int nd = enode[tt];
    const float* row = emb2 + (long)nd * 64;
    float p = 0.0f;
#pragma unroll
    for (int q = 0; q < 4; ++q)
        p += row[j + 16 * q] * sWo[j + 16 * q];
    if (j < 6) p += ef[tt * 6 + j] * sWo[64 + j];
#pragma unroll
    for (int mask = 8; mask >= 1; mask >>= 1)
        p += __shfl_xor(p, mask, 16);
    if (ok && j == 0) out[t] = p + sb;
}

extern "C" void kernel_launch(void* const* d_in, const int* in_sizes, int n_in,
                              void* d_out, int out_size, void* d_ws, size_t ws_size,
                              hipStream_t stream) {
    (void)n_in; (void)out_size; (void)ws_size;
    const float* lf   = (const float*)d_in[0];
    const float* ef   = (const float*)d_in[1];
    const float* Wa0  = (const float*)d_in[2];
    const float* ba0  = (const float*)d_in[3];
    const float* Wa1  = (const float*)d_in[4];
    const float* ba1  = (const float*)d_in[5];
    const float* Wn0  = (const float*)d_in[6];
    const float* bn0  = (const float*)d_in[7];
    const float* Wn1  = (const float*)d_in[8];
    const float* bn1  = (const float*)d_in[9];
    const float* Wo   = (const float*)d_in[10];
    const float* bo   = (const float*)d_in[11];
    const int* nbr    = (const int*)d_in[12];
    const int* srcidx = (const int*)d_in[13];
    const int* enode  = (const int*)d_in[14];

    const int E = in_sizes[12];          // 1.6M edges
    const int N = E / 16;                // DEG = 16, edges dest-sorted
    const int T = in_sizes[14];          // routing entries

    char* ws = (char*)d_ws;
    size_t off = 0;
    int*   deg  = (int*)(ws + off);   off += (size_t)N * 4;
    float* emb0 = (float*)(ws + off); off += (size_t)N * 4;
    off = (off + 255) & ~(size_t)255;
    float* agg  = (float*)(ws + off); off += (size_t)N * 256;   // reused both rounds
    float* emb1 = (float*)(ws + off); off += (size_t)N * 256;
    float* emb2 = (float*)(ws + off); off += (size_t)N * 256;

    const int B = 256;                   // 8 waves / block
    const int WPB = 8;                   // node-waves per block
    const int tiles = (N + 15) / 16;
    const int entriesPerBlock = (B / 32) * 2;

    k_zero_i<<<(N + B - 1) / B, B, 0, stream>>>(deg, N);
    k_deg  <<<(E + B - 1) / B, B, 0, stream>>>(srcidx, deg, E);
    k_emb0 <<<(N + B - 1) / B, B, 0, stream>>>(deg, emb0, N, 1.0f / (float)N);
    k_edge1<<<(N + WPB - 1) / WPB, B, 0, stream>>>(emb0, lf, nbr, Wa0, ba0, agg, N);
    k_node1<<<(tiles + WPB - 1) / WPB, B, 0, stream>>>(agg, emb0, Wn0, bn0, emb1, N);
    k_edge2<<<(N + WPB - 1) / WPB, B, 0, stream>>>(emb1, lf, nbr, Wa1, ba1, agg, N);
    k_node2<<<(tiles + WPB - 1) / WPB, B, 0, stream>>>(agg, emb1, Wn1, bn1, emb2, N);
    k_out  <<<(T + entriesPerBlock - 1) / entriesPerBlock, B, 0, stream>>>(
        emb2, ef, enode, Wo, bo, (float*)d_out, T);
}